// ATSSPostProcessor_50422916055496
// MI455X (gfx1250) — compile-verified
//
#include <hip/hip_runtime.h>
#include <hip/hip_bf16.h>

typedef float v2f __attribute__((ext_vector_type(2)));
typedef float v8f __attribute__((ext_vector_type(8)));

constexpr int N_ = 4, C_ = 15, H_ = 256, W_ = 256;
constexpr int K_ = H_ * W_;          // 65536 locations
constexpr int KC = K_ * C_;          // 983040 candidates per image
constexpr int TOPK = 400, POST = 100, ABINS = 90;
constexpr int BINS = 4096;           // score histogram bins
constexpr int CAP  = 4096;           // compaction capacity per image (32KB LDS sort)
constexpr int TILES = 25;            // 400 / 16
constexpr float PRE_TH = 0.05f, NMS_TH = 0.4f;

// ---------------- workspace layout (bytes) ----------------
constexpr size_t AL(size_t x) { return (x + 255) & ~size_t(255); }
constexpr size_t OFF_HIST = 0;                                          // u32 [N][BINS]
constexpr size_t OFF_THR  = AL(OFF_HIST + size_t(N_) * BINS * 4);       // u32 [N]
constexpr size_t OFF_CNT  = AL(OFF_THR  + size_t(N_) * 4);              // u32 [N]
constexpr size_t OFF_CMP  = AL(OFF_CNT  + size_t(N_) * 4);              // u64 [N][CAP]
constexpr size_t OFF_TOPV = AL(OFF_CMP  + size_t(N_) * CAP * 8);        // f32 [N][TOPK]
constexpr size_t OFF_TOPI = AL(OFF_TOPV + size_t(N_) * TOPK * 4);       // i32 [N][TOPK]
constexpr size_t OFF_COR  = AL(OFF_TOPI + size_t(N_) * TOPK * 4);       // f32 [N][TOPK][8]
constexpr size_t OFF_SC   = AL(OFF_COR  + size_t(N_) * TOPK * 32);      // f32 [N][TOPK]
constexpr size_t OFF_LAB  = AL(OFF_SC   + size_t(N_) * TOPK * 4);       // i32 [N][TOPK]
constexpr size_t OFF_VAL  = AL(OFF_LAB  + size_t(N_) * TOPK * 4);       // i32 [N][TOPK]
constexpr size_t OFF_CXY  = AL(OFF_VAL  + size_t(N_) * TOPK * 4);       // f32 [N][TOPK][2]
constexpr size_t OFF_NRM  = AL(OFF_CXY  + size_t(N_) * TOPK * 8);       // f32 [N][TOPK]
constexpr size_t OFF_RAD  = AL(OFF_NRM  + size_t(N_) * TOPK * 4);       // f32 [N][TOPK]
constexpr size_t OFF_AREA = AL(OFF_RAD  + size_t(N_) * TOPK * 4);       // f32 [N][TOPK]
constexpr size_t OFF_CAND = AL(OFF_AREA + size_t(N_) * TOPK * 4);       // u8  [N][TOPK][TOPK]
constexpr size_t OFF_SUP  = AL(OFF_CAND + size_t(N_) * TOPK * TOPK);    // u8  [N][TOPK][TOPK]

__device__ __forceinline__ float sigm(float x) { return 1.0f / (1.0f + __expf(-x)); }

// ---------------- K0: zero histogram + counters (ws is not re-poisoned) ----
__global__ void k_zero(unsigned* hist, unsigned* cnt) {
  int gid = blockIdx.x * blockDim.x + threadIdx.x;
  if (gid < N_ * BINS) hist[gid] = 0u;
  if (gid < N_) cnt[gid] = 0u;
}

// ---------------- K1: fused sigmoid + score histogram ----------------------
__global__ void k_hist(const float* __restrict__ cls, const float* __restrict__ ctr,
                       unsigned* __restrict__ hist) {
  int gid = blockIdx.x * blockDim.x + threadIdx.x;
  if (gid >= N_ * KC) return;
  int n = gid / KC, r = gid % KC;
  int hw = r % K_;
  float sc = sigm(cls[gid]);
  if (sc <= PRE_TH) return;
  float s = sc * sigm(ctr[n * K_ + hw]);       // comb score, in (0,1)
  unsigned bin = __float_as_uint(s) >> 18;     // order-preserving for positive floats
  if (bin >= BINS) bin = BINS - 1;
  atomicAdd(&hist[n * BINS + bin], 1u);
}

// ---------------- K2: find threshold bin (cum >= TOPK from top) ------------
__global__ void k_thresh(const unsigned* __restrict__ hist, unsigned* __restrict__ thr) {
  if (threadIdx.x != 0) return;
  int n = blockIdx.x;
  const unsigned* h = hist + n * BINS;
  unsigned cum = 0, t = 0;
  for (int b = BINS - 1; b >= 0; --b) {
    cum += h[b];
    if (cum >= (unsigned)TOPK) { t = (unsigned)b; break; }
  }
  thr[n] = t;
}

// ---------------- K3: compact candidates >= threshold bin ------------------
__global__ void k_compact(const float* __restrict__ cls, const float* __restrict__ ctr,
                          const unsigned* __restrict__ thr, unsigned* __restrict__ cnt,
                          unsigned long long* __restrict__ buf) {
  int gid = blockIdx.x * blockDim.x + threadIdx.x;
  if (gid >= N_ * KC) return;
  int n = gid / KC, r = gid % KC;
  int c = r / K_, hw = r % K_;
  float sc = sigm(cls[gid]);
  if (sc <= PRE_TH) return;
  float s = sc * sigm(ctr[n * K_ + hw]);
  unsigned bits = __float_as_uint(s);
  unsigned bin = bits >> 18; if (bin >= BINS) bin = BINS - 1;
  if (bin < thr[n]) return;
  unsigned t = (unsigned)(hw * C_ + c);        // flat index in (K,C) reshape
  unsigned p = atomicAdd(&cnt[n], 1u);
  if (p < (unsigned)CAP)                       // composite key: desc score, asc index
    buf[(size_t)n * CAP + p] = ((unsigned long long)bits << 32) | (unsigned)(~t);
}

// ---------------- K4: in-LDS bitonic sort -> exact top-400 -----------------
__global__ __launch_bounds__(1024) void k_sort(const unsigned* __restrict__ cnt,
                                               const unsigned long long* __restrict__ buf,
                                               float* __restrict__ topv, int* __restrict__ topi) {
  __shared__ unsigned long long s[CAP];
  int n = blockIdx.x, tid = threadIdx.x;
  unsigned m = cnt[n]; if (m > (unsigned)CAP) m = CAP;
  for (int i = tid; i < CAP; i += blockDim.x)
    s[i] = (i < (int)m) ? buf[(size_t)n * CAP + i] : 0ull;
  __syncthreads();
  for (int k = 2; k <= CAP; k <<= 1) {
    for (int j = k >> 1; j > 0; j >>= 1) {
      for (int i = tid; i < CAP; i += blockDim.x) {
        int ixj = i ^ j;
        if (ixj > i) {
          unsigned long long a = s[i], b = s[ixj];
          bool asc = ((i & k) == 0);
          if (asc ? (a > b) : (a < b)) { s[i] = b; s[ixj] = a; }
        }
      }
      __syncthreads();
    }
  }
  for (int t = tid; t < TOPK; t += blockDim.x) {
    unsigned long long key = s[CAP - 1 - t];   // ascending sort -> read from top
    topv[n * TOPK + t] = key ? __uint_as_float((unsigned)(key >> 32)) : 0.0f;
    topi[n * TOPK + t] = key ? (int)(~(unsigned)key) : 0;
  }
}

// ---------------- K5: gather + angle argmax + box decode + corners ---------
__global__ void k_decode(const float* __restrict__ reg, const float* __restrict__ anch,
                         const float* __restrict__ ang, const float* __restrict__ topv,
                         const int* __restrict__ topi, float* __restrict__ corners,
                         float* __restrict__ scores, int* __restrict__ labels,
                         int* __restrict__ valid, float* __restrict__ cxy,
                         float* __restrict__ nrm, float* __restrict__ rad,
                         float* __restrict__ area) {
  int gid = blockIdx.x * blockDim.x + threadIdx.x;
  if (gid >= N_ * TOPK) return;
  int n = gid / TOPK;
  float v = topv[gid];
  int idx = topi[gid];
  int loc = idx / C_, lab = idx % C_ + 1;
  // softmax is monotone -> argmax of raw angle logits
  const float* ap = ang + (size_t)n * ABINS * K_ + loc;
  float best = ap[0]; int bi = 0;
  for (int b = 1; b < ABINS; ++b) {
    float x = ap[(size_t)b * K_];
    if (x > best) { best = x; bi = b; }
  }
  float pang = (float)bi - 90.0f;
  const float* rp = reg + (size_t)n * 4 * K_ + loc;
  float dx = rp[0] * 0.1f, dy = rp[(size_t)K_] * 0.1f;
  float dw = fminf(fmaxf(rp[2 * (size_t)K_] * 0.2f, -10.f), 4.f);
  float dh = fminf(fmaxf(rp[3 * (size_t)K_] * 0.2f, -10.f), 4.f);
  const float* an = anch + ((size_t)n * K_ + loc) * 5;
  float acx = an[0], acy = an[1], aw = an[2], ah = an[3];
  float cx = dx * aw + acx, cy = dy * ah + acy;
  float w = aw * __expf(dw), h = ah * __expf(dh);
  bool ok = (v > 0.f) && (w >= 0.f) && (h >= 0.f);
  scores[gid] = ok ? sqrtf(fmaxf(v, 0.f)) : 0.f;
  labels[gid] = lab;
  valid[gid] = ok ? 1 : 0;
  float tr = pang * 0.017453292519943295f;
  float cth = __cosf(tr), sth = __sinf(tr);
  const float oxs[4] = {1.f, -1.f, -1.f, 1.f};
  const float oys[4] = {1.f, 1.f, -1.f, -1.f};
  float* co = corners + (size_t)gid * 8;
  #pragma unroll
  for (int q = 0; q < 4; ++q) {
    float ox = oxs[q] * w * 0.5f, oy = oys[q] * h * 0.5f;
    co[2 * q]     = cx + ox * cth - oy * sth;
    co[2 * q + 1] = cy + ox * sth + oy * cth;
  }
  cxy[2 * gid] = cx; cxy[2 * gid + 1] = cy;
  nrm[gid] = cx * cx + cy * cy;
  rad[gid] = 0.5f * sqrtf(w * w + h * h);      // half-diagonal
  area[gid] = w * h;
}

// ---------------- K6: WMMA center-distance prefilter -----------------------
// Per image: G = X X^T with X[400][4] = [cx, cy, 0, 0]; dist^2 = n_i + n_j - 2*G.
// Conservative reject: dist > r_i + r_j  =>  boxes cannot intersect (IoU = 0).
// One wave = one 16x16 output tile via V_WMMA_F32_16X16X4_F32.
__global__ void k_prefilter(const float* __restrict__ cxy, const float* __restrict__ nrm,
                            const float* __restrict__ rad, unsigned char* __restrict__ cand) {
  int bid = blockIdx.x;
  int n = bid / (TILES * TILES), t = bid % (TILES * TILES);
  int i0 = (t / TILES) * 16, j0 = (t % TILES) * 16;
  int lane = threadIdx.x;
  int m = lane & 15;
  bool hi = lane >= 16;
  const float* X = cxy + (size_t)n * TOPK * 2;
  // A (16x4): lanes 0-15 carry K={0,1}=(cx,cy) for row m; lanes 16-31 carry K={2,3}=0
  v2f a; a.x = hi ? 0.f : X[2 * (i0 + m)];
         a.y = hi ? 0.f : X[2 * (i0 + m) + 1];
  // B (4x16): VGPR0 = rows K0/K1 (cx_j / cy_j) striped across lane halves; VGPR1 = K2/K3 = 0
  v2f b; b.x = hi ? X[2 * (j0 + m) + 1] : X[2 * (j0 + m)];
         b.y = 0.f;
  v8f c = {};
  c = __builtin_amdgcn_wmma_f32_16x16x4_f32(false, a, false, b, (short)0, c, false, false);
  const float* pN = nrm + (size_t)n * TOPK;
  const float* pR = rad + (size_t)n * TOPK;
  unsigned char* pM = cand + (size_t)n * TOPK * TOPK;
  #pragma unroll
  for (int v = 0; v < 8; ++v) {                // D layout: lane<16 -> M=v ; lane>=16 -> M=v+8
    int i = i0 + (hi ? v + 8 : v);
    int j = j0 + m;
    float d2 = pN[i] + pN[j] - 2.f * c[v];
    float rr = pR[i] + pR[j];
    pM[(size_t)i * TOPK + j] = (d2 <= rr * rr + 1.0f) ? 1 : 0;
  }
}

// ---------------- K7: rotated IoU (upper triangle) -> suppression mask -----
__global__ void k_iou(const float* __restrict__ corners, const float* __restrict__ area,
                      const int* __restrict__ labels, const int* __restrict__ valid,
                      const unsigned char* __restrict__ cand, unsigned char* __restrict__ sup) {
  int gid = blockIdx.x * blockDim.x + threadIdx.x;
  if (gid >= N_ * TOPK * TOPK) return;
  int n = gid / (TOPK * TOPK), r = gid % (TOPK * TOPK);
  int i = r / TOPK, j = r % TOPK;
  unsigned char out = 0;
  if (j > i && valid[n * TOPK + i] && valid[n * TOPK + j] &&
      labels[n * TOPK + i] == labels[n * TOPK + j] && cand[gid]) {
    const float* ci = corners + (size_t)(n * TOPK + i) * 8;
    const float* cj = corners + (size_t)(n * TOPK + j) * 8;
    float px[16], py[16], qx[16], qy[16];
    int nv = 4;
    #pragma unroll
    for (int v = 0; v < 4; ++v) { px[v] = ci[2 * v]; py[v] = ci[2 * v + 1]; }
    // Sutherland-Hodgman: clip quad i against the 4 CCW edges of quad j
    for (int e = 0; e < 4 && nv > 0; ++e) {
      float x1 = cj[2 * e], y1 = cj[2 * e + 1];
      int e2 = (e + 1) & 3;
      float ex = cj[2 * e2] - x1, ey = cj[2 * e2 + 1] - y1;
      int mc = 0;
      for (int v = 0; v < nv; ++v) {
        int w2 = (v + 1 == nv) ? 0 : v + 1;
        float dc = ex * (py[v] - y1) - ey * (px[v] - x1);
        float dn = ex * (py[w2] - y1) - ey * (px[w2] - x1);
        if (dc >= 0.f && mc < 16) { qx[mc] = px[v]; qy[mc] = py[v]; mc++; }
        if (((dc >= 0.f) != (dn >= 0.f)) && mc < 16) {
          float den = dc - dn;
          if (fabsf(den) < 1e-8f) den = 1e-8f;
          float tp = dc / den;
          qx[mc] = px[v] + tp * (px[w2] - px[v]);
          qy[mc] = py[v] + tp * (py[w2] - py[v]);
          mc++;
        }
      }
      nv = mc;
      for (int v = 0; v < nv; ++v) { px[v] = qx[v]; py[v] = qy[v]; }
    }
    float ss = 0.f;
    for (int v = 0; v < nv; ++v) {
      int w2 = (v + 1 == nv) ? 0 : v + 1;
      ss += px[v] * py[w2] - px[w2] * py[v];
    }
    float inter = 0.5f * fabsf(ss);
    float iou = inter / (area[n * TOPK + i] + area[n * TOPK + j] - inter + 1e-7f);
    out = (iou > NMS_TH) ? 1 : 0;
  }
  sup[gid] = out;
}

// ---------------- K8: greedy NMS + final top-100 + output ------------------
__global__ __launch_bounds__(512) void k_nms_out(const unsigned char* __restrict__ sup,
                                                 const int* __restrict__ valid,
                                                 const float* __restrict__ scores,
                                                 const int* __restrict__ labels,
                                                 const float* __restrict__ corners,
                                                 float* __restrict__ out) {
  __shared__ int keep[TOPK];
  int n = blockIdx.x, tid = threadIdx.x;
  for (int i = tid; i < TOPK; i += blockDim.x) keep[i] = valid[n * TOPK + i];
  const unsigned char* S = sup + (size_t)n * TOPK * TOPK;
  for (int i = 0; i < TOPK; ++i) {
    __syncthreads();
    if (keep[i]) {
      for (int j = i + 1 + tid; j < TOPK; j += blockDim.x)
        if (S[(size_t)i * TOPK + j]) keep[j] = 0;
    }
  }
  __syncthreads();
  if (tid == 0) {
    // scores are sorted descending already; top-100 = first 100 kept positives
    float* o = out + (size_t)n * POST * 11;
    int cnt = 0;
    for (int i = 0; i < TOPK && cnt < POST; ++i) {
      float sc = scores[n * TOPK + i];
      if (keep[i] && sc > 0.f) {
        const float* co = corners + (size_t)(n * TOPK + i) * 8;
        float* row = o + cnt * 11;
        for (int q = 0; q < 8; ++q) row[q] = co[q];
        row[8] = sc;
        row[9] = (float)labels[n * TOPK + i];
        row[10] = 1.f;
        cnt++;
      }
    }
    for (; cnt < POST; ++cnt) {
      float* row = o + cnt * 11;
      for (int q = 0; q < 11; ++q) row[q] = 0.f;
    }
  }
}

extern "C" void kernel_launch(void* const* d_in, const int* in_sizes, int n_in,
                              void* d_out, int out_size, void* d_ws, size_t ws_size,
                              hipStream_t stream) {
  const float* cls  = (const float*)d_in[0];   // [4,15,256,256]
  const float* reg  = (const float*)d_in[1];   // [4,4,256,256]
  const float* ctr  = (const float*)d_in[2];   // [4,1,256,256]
  const float* ang  = (const float*)d_in[3];   // [4,90,256,256]
  const float* anch = (const float*)d_in[4];   // [4,65536,5]
  float* out = (float*)d_out;                  // [4,100,11]
  char* ws = (char*)d_ws;

  unsigned*           hist = (unsigned*)(ws + OFF_HIST);
  unsigned*           thr  = (unsigned*)(ws + OFF_THR);
  unsigned*           cnt  = (unsigned*)(ws + OFF_CNT);
  unsigned long long* cmp  = (unsigned long long*)(ws + OFF_CMP);
  float*              topv = (float*)(ws + OFF_TOPV);
  int*                topi = (int*)(ws + OFF_TOPI);
  float*              cor  = (float*)(ws + OFF_COR);
  float*              sc   = (float*)(ws + OFF_SC);
  int*                lab  = (int*)(ws + OFF_LAB);
  int*                val  = (int*)(ws + OFF_VAL);
  float*              cxy  = (float*)(ws + OFF_CXY);
  float*              nrm  = (float*)(ws + OFF_NRM);
  float*              rad  = (float*)(ws + OFF_RAD);
  float*              area = (float*)(ws + OFF_AREA);
  unsigned char*      cand = (unsigned char*)(ws + OFF_CAND);
  unsigned char*      sup  = (unsigned char*)(ws + OFF_SUP);

  k_zero<<<(N_ * BINS + 255) / 256, 256, 0, stream>>>(hist, cnt);
  k_hist<<<(N_ * KC + 255) / 256, 256, 0, stream>>>(cls, ctr, hist);
  k_thresh<<<N_, 32, 0, stream>>>(hist, thr);
  k_compact<<<(N_ * KC + 255) / 256, 256, 0, stream>>>(cls, ctr, thr, cnt, cmp);
  k_sort<<<N_, 1024, 0, stream>>>(cnt, cmp, topv, topi);
  k_decode<<<(N_ * TOPK + 127) / 128, 128, 0, stream>>>(reg, anch, ang, topv, topi,
                                                        cor, sc, lab, val, cxy, nrm, rad, area);
  k_prefilter<<<N_ * TILES * TILES, 32, 0, stream>>>(cxy, nrm, rad, cand);
  k_iou<<<(N_ * TOPK * TOPK + 255) / 256, 256, 0, stream>>>(cor, area, lab, val, cand, sup);
  k_nms_out<<<N_, 512, 0, stream>>>(sup, val, sc, lab, cor, out);
}